// DiffNet_50620484550928
// MI455X (gfx1250) — compile-verified
//
#include <hip/hip_runtime.h>

typedef float v2f __attribute__((ext_vector_type(2)));
typedef float v8f __attribute__((ext_vector_type(8)));

#define CH 32
#define PADH 261
#define PADW 274
#define NB 16
#define ZBATCH ((size_t)PADH * PADW * CH)
#define ZBUF_ELEMS ((size_t)NB * ZBATCH)
#define NPIX (16 * 256 * 256)
#define STATS_BLOCKS 1024

__device__ __forceinline__ size_t zoff(int b, int h, int w) {
  return (size_t)b * ZBATCH + ((size_t)h * PADW + (size_t)w) * CH;
}

// ---------------- conv1: 1->32 channels, scalar (1.3% of FLOPs) ----------------
__global__ void conv1_kernel(const float* __restrict__ x, const float* __restrict__ w1,
                             const float* __restrict__ b1, float* __restrict__ y, int layer) {
  int idx = blockIdx.x * blockDim.x + threadIdx.x;
  if (idx >= NPIX) return;
  int w = idx & 255, h = (idx >> 8) & 255, b = idx >> 16;
  const float* wp = w1 + (size_t)layer * CH * 9;
  float xv[9];
#pragma unroll
  for (int dy = 0; dy < 3; ++dy)
#pragma unroll
    for (int dx = 0; dx < 3; ++dx) {
      int hh = h + dy - 1, ww = w + dx - 1;
      xv[dy * 3 + dx] = (hh >= 0 && hh < 256 && ww >= 0 && ww < 256)
                            ? x[((size_t)b * 256 + hh) * 256 + ww] : 0.f;
    }
  float* yo = y + zoff(b, h + 1, w + 1);
#pragma unroll 4
  for (int c = 0; c < CH; ++c) {
    float acc = b1[layer * CH + c];
#pragma unroll
    for (int t = 0; t < 9; ++t) acc += xv[t] * wp[c * 9 + t];
    yo[c] = acc;
  }
}

// ---------------- weight repack into per-lane WMMA B-fragment order ----------------
// wsrc: (L,32,32,3,3) OIHW -> wp: (L,2,72,32,2); k = (dy*3+dx)*32 + cin
__global__ void pack32_kernel(const float* __restrict__ wsrc, float* __restrict__ wp) {
  int idx = blockIdx.x * blockDim.x + threadIdx.x;
  if (idx >= 5 * 9216) return;
  int e = idx & 1;
  int lane = (idx >> 1) & 31;
  int q = (idx >> 6) % 72;
  int half = ((idx >> 6) / 72) & 1;
  int layer = idx / 9216;
  int n = (lane & 15) + 16 * half;
  int k = q * 4 + ((lane >> 4) << 1) + e;
  int cin = k & 31, tap = k >> 5;
  wp[idx] = wsrc[(((size_t)layer * 32 + n) * 32 + cin) * 9 + tap];
}

// wsrc: (L,5,32,3,3) -> wp: (L,72,32,2), N padded to 16 with zeros
__global__ void pack16_kernel(const float* __restrict__ wsrc, float* __restrict__ wp) {
  int idx = blockIdx.x * blockDim.x + threadIdx.x;
  if (idx >= 5 * 4608) return;
  int e = idx & 1;
  int lane = (idx >> 1) & 31;
  int q = (idx >> 6) % 72;
  int layer = idx / 4608;
  int n = lane & 15;
  int k = q * 4 + ((lane >> 4) << 1) + e;
  int cin = k & 31, tap = k >> 5;
  wp[idx] = (n < 5) ? wsrc[(((size_t)layer * 5 + n) * 32 + cin) * 9 + tap] : 0.f;
}

// ---------------- implicit-GEMM conv 32->32 via V_WMMA_F32_16X16X4_F32 ----------------
__global__ void __launch_bounds__(256) conv_wmma32_kernel(
    const float* __restrict__ zin, const float* __restrict__ wp,
    const float* __restrict__ bias, float* __restrict__ y,
    int Hc, int Wc, int NT, int layer) {
  __shared__ __align__(16) float ldsw[9216];
  const float* wsrc = wp + (size_t)layer * 9216;
  for (int i = threadIdx.x; i < 9216; i += blockDim.x) ldsw[i] = wsrc[i];
  __syncthreads();

  int tile = blockIdx.x * (blockDim.x >> 5) + (threadIdx.x >> 5);
  int totalTiles = NB * Hc * NT;
  if (tile >= totalTiles) return;
  int wt = tile % NT;
  int h  = (tile / NT) % Hc;
  int b  = tile / (NT * Hc);

  int lane = threadIdx.x & 31;
  int m    = lane & 15;              // pixel within tile (A-frag M)
  int kl   = (lane >> 4) << 1;       // K sub-offset 0 or 2
  int wpix = wt * 16 + m;

  const v2f* lw = (const v2f*)ldsw;
  v8f acc0 = {}; v8f acc1 = {};
  const float* abase = zin + zoff(b, h, wpix) + kl;
#pragma unroll
  for (int dy = 0; dy < 3; ++dy) {
#pragma unroll
    for (int dx = 0; dx < 3; ++dx) {
      const float* ap = abase + ((size_t)dy * PADW + dx) * CH;
      int qb = (dy * 3 + dx) * 8;
#pragma unroll
      for (int cc = 0; cc < 8; ++cc) {
        v2f a  = *(const v2f*)(ap + cc * 4);
        v2f b0 = lw[(qb + cc) * 32 + lane];
        v2f b1 = lw[(72 + qb + cc) * 32 + lane];
        acc0 = __builtin_amdgcn_wmma_f32_16x16x4_f32(false, a, false, b0, (short)0, acc0, false, false);
        acc1 = __builtin_amdgcn_wmma_f32_16x16x4_f32(false, a, false, b1, (short)0, acc1, false, false);
      }
    }
  }
  int n0 = lane & 15;
  float bi0 = bias[layer * 32 + n0];
  float bi1 = bias[layer * 32 + n0 + 16];
  int mbase = (lane >> 4) << 3;      // C/D frag: lanes 16-31 hold rows r+8
#pragma unroll
  for (int r = 0; r < 8; ++r) {
    int wr = wt * 16 + mbase + r;
    if (wr < Wc) {
      size_t o = zoff(b, h + 1, wr + 1);
      y[o + n0]      = acc0[r] + bi0;
      y[o + n0 + 16] = acc1[r] + bi1;
    }
  }
}

// ---------------- conv4 (32->5, N padded to 16) + bias + relu + center-crop -> meta ----------------
__global__ void __launch_bounds__(256) conv_meta_kernel(
    const float* __restrict__ zin, const float* __restrict__ wp4,
    const float* __restrict__ b4, float* __restrict__ meta, int layer) {
  __shared__ __align__(16) float ldsw[4608];
  const float* wsrc = wp4 + (size_t)layer * 4608;
  for (int i = threadIdx.x; i < 4608; i += blockDim.x) ldsw[i] = wsrc[i];
  __syncthreads();

  int tile = blockIdx.x * (blockDim.x >> 5) + (threadIdx.x >> 5);
  if (tile >= NB * 256 * 16) return;
  int wt = tile & 15;
  int h  = (tile >> 4) & 255;
  int b  = tile >> 12;

  int lane = threadIdx.x & 31;
  int m    = lane & 15;
  int kl   = (lane >> 4) << 1;
  int wpix = wt * 16 + m;

  const v2f* lw = (const v2f*)ldsw;
  v8f acc = {};
  // crop offset 1: conv output coord = (h+1, wpix+1)
  const float* abase = zin + zoff(b, h + 1, wpix + 1) + kl;
#pragma unroll
  for (int dy = 0; dy < 3; ++dy)
#pragma unroll
    for (int dx = 0; dx < 3; ++dx) {
      const float* ap = abase + ((size_t)dy * PADW + dx) * CH;
      int qb = (dy * 3 + dx) * 8;
#pragma unroll
      for (int cc = 0; cc < 8; ++cc) {
        v2f a  = *(const v2f*)(ap + cc * 4);
        v2f bb = lw[(qb + cc) * 32 + lane];
        acc = __builtin_amdgcn_wmma_f32_16x16x4_f32(false, a, false, bb, (short)0, acc, false, false);
      }
    }
  int n = lane & 15;
  if (n < 5) {
    float bi = b4[layer * 5 + n];
    int mbase = (lane >> 4) << 3;
#pragma unroll
    for (int r = 0; r < 8; ++r) {
      int w = wt * 16 + mbase + r;
      meta[(((size_t)b * 256 + h) * 256 + w) * 5 + n] = fmaxf(acc[r] + bi, 0.f);
    }
  }
}

// ---------------- BN stats: deterministic two-stage reduction ----------------
__global__ void stats_kernel(const float* __restrict__ y, float* __restrict__ part,
                             int Hc, int Wc) {
  int tid = threadIdx.x;
  int c = tid & 31;
  int npix = NB * Hc * Wc;
  int ngroups = (gridDim.x * blockDim.x) >> 5;
  int p0 = (blockIdx.x * blockDim.x + tid) >> 5;
  float s = 0.f, q = 0.f;
  for (int p = p0; p < npix; p += ngroups) {
    int w = p % Wc; int t = p / Wc; int h = t % Hc; int b = t / Hc;
    float v = y[zoff(b, h + 1, w + 1) + c];
    s += v; q += v * v;
  }
  __shared__ float ls[256], lq[256];
  ls[tid] = s; lq[tid] = q;
  __syncthreads();
  for (int ofs = 128; ofs >= 32; ofs >>= 1) {
    if (tid < ofs) { ls[tid] += ls[tid + ofs]; lq[tid] += lq[tid + ofs]; }
    __syncthreads();
  }
  if (tid < 32) {
    part[(size_t)blockIdx.x * 64 + tid]      = ls[tid];
    part[(size_t)blockIdx.x * 64 + 32 + tid] = lq[tid];
  }
}

__global__ void stats_reduce_kernel(const float* __restrict__ part,
                                    const float* __restrict__ g, const float* __restrict__ be,
                                    float* __restrict__ sc, int layer, float invcnt, int nblocks) {
  int t = threadIdx.x;  // 32 threads; channel t
  if (t >= 32) return;
  float s = 0.f, q = 0.f;
  for (int i = 0; i < nblocks; ++i) {
    s += part[(size_t)i * 64 + t];
    q += part[(size_t)i * 64 + 32 + t];
  }
  float mean = s * invcnt;
  float var  = q * invcnt - mean * mean;
  float istd = rsqrtf(var + 1e-5f);
  float scale = g[layer * 32 + t] * istd;
  sc[t]      = scale;
  sc[32 + t] = be[layer * 32 + t] - mean * scale;
}

// ---------------- fused BN + ReLU + overlapping maxpool; zero-fills halo ----------------
__global__ void bnpool_kernel(const float* __restrict__ y, const float* __restrict__ sc,
                              float* __restrict__ z, int Hc, int Wc) {
  size_t idx = (size_t)blockIdx.x * blockDim.x + threadIdx.x;
  if (idx >= ZBUF_ELEMS) return;
  int c = (int)(idx & 31);
  size_t p = idx >> 5;
  int wb = (int)(p % PADW);
  size_t t = p / PADW;
  int hb = (int)(t % PADH);
  int b  = (int)(t / PADH);
  float out = 0.f;
  int Po = Hc + 1, PoW = Wc + 1;
  if (hb >= 1 && hb <= Po && wb >= 1 && wb <= PoW) {
    int oh = hb - 1, ow = wb - 1;
    float scale = sc[c], shift = sc[32 + c];
    float mx = -INFINITY;
#pragma unroll
    for (int dy = -1; dy <= 0; ++dy)
#pragma unroll
      for (int dx = -1; dx <= 0; ++dx) {
        int h2 = oh + dy, w2 = ow + dx;
        if (h2 >= 0 && h2 < Hc && w2 >= 0 && w2 < Wc) {
          float v = y[zoff(b, h2 + 1, w2 + 1) + c];
          mx = fmaxf(mx, v * scale + shift);
        }
      }
    out = fmaxf(mx, 0.f);  // relu commutes with max
  }
  z[idx] = out;
}

// ---------------- x stencil update ----------------
__global__ void update_kernel(const float* __restrict__ x, const float* __restrict__ meta,
                              const float* __restrict__ dtp, float* __restrict__ xn) {
  int idx = blockIdx.x * blockDim.x + threadIdx.x;
  if (idx >= NPIX) return;
  int w = idx & 255, h = (idx >> 8) & 255;
  float dt = *dtp;
  const float* mp = meta + (size_t)idx * 5;
  float xc = x[idx];
  float up = (h > 0)   ? x[idx - 256] : 0.f;
  float dn = (h < 255) ? x[idx + 256] : 0.f;
  float lf = (w < 255) ? x[idx + 1]   : 0.f;
  float rt = (w > 0)   ? x[idx - 1]   : 0.f;
  float upd = -xc * mp[0] + up * mp[1] + dn * mp[2] + lf * mp[3] + rt * mp[4];
  xn[idx] = xc - dt * upd;
}

__global__ void relu_kernel(const float* __restrict__ x, float* __restrict__ out, int n) {
  int i = blockIdx.x * blockDim.x + threadIdx.x;
  if (i < n) out[i] = fmaxf(x[i], 0.f);
}

// ---------------- host launcher ----------------
extern "C" void kernel_launch(void* const* d_in, const int* in_sizes, int n_in,
                              void* d_out, int out_size, void* d_ws, size_t ws_size,
                              hipStream_t stream) {
  const float* x0  = (const float*)d_in[0];
  const float* w1  = (const float*)d_in[1];
  const float* b1  = (const float*)d_in[2];
  const float* g1  = (const float*)d_in[3];
  const float* be1 = (const float*)d_in[4];
  const float* w2  = (const float*)d_in[5];
  const float* b2  = (const float*)d_in[6];
  const float* g2  = (const float*)d_in[7];
  const float* be2 = (const float*)d_in[8];
  const float* w3  = (const float*)d_in[9];
  const float* b3  = (const float*)d_in[10];
  const float* g3  = (const float*)d_in[11];
  const float* be3 = (const float*)d_in[12];
  const float* w4  = (const float*)d_in[13];
  const float* b4  = (const float*)d_in[14];
  const float* dt  = (const float*)d_in[15];

  float* ws = (float*)d_ws;
  size_t off = 0;
  float* zA   = ws + off; off += ZBUF_ELEMS;
  float* zB   = ws + off; off += ZBUF_ELEMS;
  float* yb   = ws + off; off += ZBUF_ELEMS;
  float* meta = ws + off; off += (size_t)NPIX * 5;
  float* xA   = ws + off; off += NPIX;
  float* xB   = ws + off; off += NPIX;
  float* wp2  = ws + off; off += 5 * 9216;
  float* wp3  = ws + off; off += 5 * 9216;
  float* wp4  = ws + off; off += 5 * 4608;
  float* part = ws + off; off += (size_t)STATS_BLOCKS * 64;
  float* sc   = ws + off; off += 64;
  if (ws_size < off * sizeof(float)) return;

  pack32_kernel<<<(5 * 9216 + 255) / 256, 256, 0, stream>>>(w2, wp2);
  pack32_kernel<<<(5 * 9216 + 255) / 256, 256, 0, stream>>>(w3, wp3);
  pack16_kernel<<<(5 * 4608 + 255) / 256, 256, 0, stream>>>(w4, wp4);

  unsigned zgrid = (unsigned)((ZBUF_ELEMS + 255) / 256);
  const float* xin = x0;
  float* xout = xA;
  for (int l = 0; l < 5; ++l) {
    // layer 1: conv1 -> yb (interior 256x256) -> bn/relu/pool -> zA (interior 257)
    conv1_kernel<<<(NPIX + 255) / 256, 256, 0, stream>>>(xin, w1, b1, yb, l);
    stats_kernel<<<STATS_BLOCKS, 256, 0, stream>>>(yb, part, 256, 256);
    stats_reduce_kernel<<<1, 32, 0, stream>>>(part, g1, be1, sc, l,
                                              1.f / (16.f * 256 * 256), STATS_BLOCKS);
    bnpool_kernel<<<zgrid, 256, 0, stream>>>(yb, sc, zA, 256, 256);

    // layer 2: WMMA conv zA(257) -> yb -> zB (interior 258)
    {
      int Hc = 257, Wc = 257, NT = 17;
      int tiles = NB * Hc * NT;
      conv_wmma32_kernel<<<(tiles + 7) / 8, 256, 0, stream>>>(zA, wp2, b2, yb, Hc, Wc, NT, l);
      stats_kernel<<<STATS_BLOCKS, 256, 0, stream>>>(yb, part, Hc, Wc);
      stats_reduce_kernel<<<1, 32, 0, stream>>>(part, g2, be2, sc, l,
                                                1.f / (16.f * Hc * Wc), STATS_BLOCKS);
      bnpool_kernel<<<zgrid, 256, 0, stream>>>(yb, sc, zB, Hc, Wc);
    }
    // layer 3: WMMA conv zB(258) -> yb -> zA (interior 259)
    {
      int Hc = 258, Wc = 258, NT = 17;
      int tiles = NB * Hc * NT;
      conv_wmma32_kernel<<<(tiles + 7) / 8, 256, 0, stream>>>(zB, wp3, b3, yb, Hc, Wc, NT, l);
      stats_kernel<<<STATS_BLOCKS, 256, 0, stream>>>(yb, part, Hc, Wc);
      stats_reduce_kernel<<<1, 32, 0, stream>>>(part, g3, be3, sc, l,
                                                1.f / (16.f * Hc * Wc), STATS_BLOCKS);
      bnpool_kernel<<<zgrid, 256, 0, stream>>>(yb, sc, zA, Hc, Wc);
    }
    // layer 4: WMMA conv (crop fused) zA(259) -> meta (B,256,256,5)
    {
      int tiles = NB * 256 * 16;
      conv_meta_kernel<<<(tiles + 7) / 8, 256, 0, stream>>>(zA, wp4, b4, meta, l);
    }
    update_kernel<<<(NPIX + 255) / 256, 256, 0, stream>>>(xin, meta, dt, xout);
    xin = xout;
    xout = (xout == xA) ? xB : xA;
  }
  relu_kernel<<<(NPIX + 255) / 256, 256, 0, stream>>>(xin, (float*)d_out, NPIX);
}